// GAT_32908039422439
// MI455X (gfx1250) — compile-verified
//
#include <hip/hip_runtime.h>
#include <math.h>

#define N_NODES   50000
#define N_EDGES   800000
#define E_TOT     850000          // + self loops
#define FDIM      128
#define N_GRAPHS  500
#define NEG_SLOPE 0.2f

typedef float v2f __attribute__((ext_vector_type(2)));
typedef float v8f __attribute__((ext_vector_type(8)));

// ---------------------------------------------------------------------------
// GEMM: H[50000x128] = maybe_relu(X)[50000x128] @ W[128x128]  via WMMA f32
// block = 256 threads = 8 waves; block owns 16 M-rows, wave owns 16 N-cols.
// ---------------------------------------------------------------------------
__global__ __launch_bounds__(256)
void gemm_wmma(const float* __restrict__ X, const float* __restrict__ W,
               float* __restrict__ H, int apply_relu_in)
{
    __shared__ float As[16][FDIM + 4];   // +4 pad: row stride 132 -> conflict free
    const int tid = threadIdx.x;
    const int M0  = blockIdx.x * 16;

    // cooperative stage of 16x128 input tile; fuse ReLU of previous layer
    #pragma unroll
    for (int i = 0; i < 8; ++i) {
        int idx = tid + i * 256;
        int r = idx >> 7, c = idx & 127;
        float v = X[(size_t)(M0 + r) * FDIM + c];
        if (apply_relu_in) v = fmaxf(v, 0.0f);
        As[r][c] = v;
    }
    __syncthreads();

    const int wave = tid >> 5;
    const int lane = tid & 31;
    const int half = lane >> 4;    // 0: lanes 0-15, 1: lanes 16-31
    const int l16  = lane & 15;
    const int N0   = wave * 16;

    v8f acc = {0.f, 0.f, 0.f, 0.f, 0.f, 0.f, 0.f, 0.f};
    for (int k = 0; k < FDIM; k += 4) {
        const int kk = k + half * 2;           // K split lo/hi across lane halves
        v2f a, b;
        // A 16x4 layout: lane l16 holds row M=l16; VGPR0=K kk, VGPR1=K kk+1
        a.x = As[l16][kk];
        a.y = As[l16][kk + 1];
        // B 4x16 layout: lane l16 holds col N0+l16; VGPR0=K kk, VGPR1=K kk+1
        b.x = W[(size_t)kk * FDIM + N0 + l16];
        b.y = W[(size_t)(kk + 1) * FDIM + N0 + l16];
        acc = __builtin_amdgcn_wmma_f32_16x16x4_f32(false, a, false, b,
                                                    (short)0, acc, false, false);
    }
    // C/D layout: VGPR r -> M = half*8 + r, N = N0 + l16
    #pragma unroll
    for (int r = 0; r < 8; ++r)
        H[(size_t)(M0 + half * 8 + r) * FDIM + N0 + l16] = acc[r];
}

// ---------------------------------------------------------------------------
// Per-node attention logits: alpha_src[n] = h[n,:]·a_src ; alpha_dst likewise.
// One wave per node, lane-strided dot + shuffle reduction.
// ---------------------------------------------------------------------------
__global__ __launch_bounds__(256)
void alpha_dots(const float* __restrict__ H,
                const float* __restrict__ a_src, const float* __restrict__ a_dst,
                float* __restrict__ out_src, float* __restrict__ out_dst)
{
    const int gtid = blockIdx.x * blockDim.x + threadIdx.x;
    const int node = gtid >> 5;
    const int lane = gtid & 31;
    if (node >= N_NODES) return;
    const float* row = H + (size_t)node * FDIM;
    float s = 0.f, d = 0.f;
    #pragma unroll
    for (int i = lane; i < FDIM; i += 32) {
        float h = row[i];
        s += h * a_src[i];
        d += h * a_dst[i];
    }
    #pragma unroll
    for (int off = 16; off; off >>= 1) {
        s += __shfl_down(s, off, 32);
        d += __shfl_down(d, off, 32);
    }
    if (lane == 0) { out_src[node] = s; out_dst[node] = d; }
}

__global__ __launch_bounds__(256)
void fill_f32(float* p, float v, int n)
{
    int i = blockIdx.x * blockDim.x + threadIdx.x;
    if (i < n) p[i] = v;
}

// agg[n][c] = bias[c]  (pre-seed aggregation buffer so atomics land on bias)
__global__ __launch_bounds__(256)
void bias_init(float* __restrict__ agg, const float* __restrict__ b)
{
    int i = blockIdx.x * blockDim.x + threadIdx.x;
    if (i < N_NODES * FDIM) agg[i] = b[i & (FDIM - 1)];
}

__device__ inline void atomicMaxF(float* addr, float val)
{
    if (val >= 0.0f) atomicMax((int*)addr, __float_as_int(val));
    else             atomicMin((unsigned int*)addr, (unsigned int)__float_as_int(val));
}

__device__ inline void edge_endpoints(const long long* __restrict__ src,
                                      const long long* __restrict__ dst,
                                      int e, int& s, int& d)
{
    if (e < N_EDGES) { s = (int)src[e]; d = (int)dst[e]; }
    else             { s = d = e - N_EDGES; }      // self loops
}

// pass 1: e = leaky_relu(asrc[src] + adst[dst]); segment max into m[dst]
__global__ __launch_bounds__(256)
void edge_score(const long long* __restrict__ src, const long long* __restrict__ dst,
                const float* __restrict__ asrc, const float* __restrict__ adst,
                float* __restrict__ escore, float* __restrict__ m)
{
    int e = blockIdx.x * blockDim.x + threadIdx.x;
    if (e >= E_TOT) return;
    int s, d; edge_endpoints(src, dst, e, s, d);
    float sc = asrc[s] + adst[d];
    sc = (sc > 0.f) ? sc : NEG_SLOPE * sc;
    escore[e] = sc;
    atomicMaxF(m + d, sc);
}

// pass 2: ex = exp(e - m[dst]); segment sum into denom[dst]
__global__ __launch_bounds__(256)
void edge_exp(const long long* __restrict__ src, const long long* __restrict__ dst,
              float* __restrict__ escore, const float* __restrict__ m,
              float* __restrict__ denom)
{
    int e = blockIdx.x * blockDim.x + threadIdx.x;
    if (e >= E_TOT) return;
    int s, d; edge_endpoints(src, dst, e, s, d);
    float ex = __expf(escore[e] - m[d]);
    escore[e] = ex;
    atomicAdd(denom + d, ex);
}

// pass 3: agg[dst,:] += h[src,:] * (ex/denom[dst]); 32 threads per edge (4 cols each)
__global__ __launch_bounds__(256)
void edge_agg(const long long* __restrict__ src, const long long* __restrict__ dst,
              const float* __restrict__ H, const float* __restrict__ ex,
              const float* __restrict__ denom, float* __restrict__ agg)
{
    long long t = (long long)blockIdx.x * blockDim.x + threadIdx.x;
    if (t >= (long long)E_TOT * 32) return;
    int e = (int)(t >> 5);
    int q = (int)(t & 31);
    int s, d; edge_endpoints(src, dst, e, s, d);
    float alpha = ex[e] / denom[d];
    float4 v = ((const float4*)(H + (size_t)s * FDIM))[q];
    float* o = agg + (size_t)d * FDIM + q * 4;
    atomicAdd(o + 0, v.x * alpha);
    atomicAdd(o + 1, v.y * alpha);
    atomicAdd(o + 2, v.z * alpha);
    atomicAdd(o + 3, v.w * alpha);
}

// mean-pool accumulation: 32 threads per node
__global__ __launch_bounds__(256)
void pool_sum(const long long* __restrict__ batch, const float* __restrict__ H,
              float* __restrict__ pooled, float* __restrict__ counts)
{
    long long t = (long long)blockIdx.x * blockDim.x + threadIdx.x;
    if (t >= (long long)N_NODES * 32) return;
    int n = (int)(t >> 5);
    int q = (int)(t & 31);
    int g = (int)batch[n];
    float4 v = ((const float4*)(H + (size_t)n * FDIM))[q];
    float* o = pooled + (size_t)g * FDIM + q * 4;
    atomicAdd(o + 0, v.x);
    atomicAdd(o + 1, v.y);
    atomicAdd(o + 2, v.z);
    atomicAdd(o + 3, v.w);
    if (q == 0) atomicAdd(counts + g, 1.0f);
}

// out[g,:] = (pooled[g,:]/max(count,1)) @ W_lin + b_lin ; one wave per graph
__global__ __launch_bounds__(256)
void final_linear(const float* __restrict__ pooled, const float* __restrict__ counts,
                  const float* __restrict__ Wl, const float* __restrict__ bl,
                  float* __restrict__ out)
{
    int t = blockIdx.x * blockDim.x + threadIdx.x;
    int g = t >> 5, lane = t & 31;
    if (g >= N_GRAPHS) return;
    float inv = 1.0f / fmaxf(counts[g], 1.0f);
    float s0 = 0.f, s1 = 0.f;
    #pragma unroll
    for (int i = lane; i < FDIM; i += 32) {
        float p = pooled[(size_t)g * FDIM + i] * inv;
        s0 += p * Wl[i * 2 + 0];
        s1 += p * Wl[i * 2 + 1];
    }
    #pragma unroll
    for (int off = 16; off; off >>= 1) {
        s0 += __shfl_down(s0, off, 32);
        s1 += __shfl_down(s1, off, 32);
    }
    if (lane == 0) {
        out[g * 2 + 0] = s0 + bl[0];
        out[g * 2 + 1] = s1 + bl[1];
    }
}

static inline size_t align_up(size_t x, size_t a) { return (x + a - 1) & ~(a - 1); }

extern "C" void kernel_launch(void* const* d_in, const int* in_sizes, int n_in,
                              void* d_out, int out_size, void* d_ws, size_t ws_size,
                              hipStream_t stream)
{
    (void)in_sizes; (void)n_in; (void)out_size; (void)ws_size;

    const float*     x     = (const float*)d_in[0];
    const long long* ei    = (const long long*)d_in[1];   // [2, 800000] int64
    const long long* batch = (const long long*)d_in[2];   // [50000] int64
    const long long* esrc  = ei;
    const long long* edst  = ei + N_EDGES;

    const float* Wm[3]  = {(const float*)d_in[3],  (const float*)d_in[7],  (const float*)d_in[11]};
    const float* avs[3] = {(const float*)d_in[4],  (const float*)d_in[8],  (const float*)d_in[12]};
    const float* avd[3] = {(const float*)d_in[5],  (const float*)d_in[9],  (const float*)d_in[13]};
    const float* bv[3]  = {(const float*)d_in[6],  (const float*)d_in[10], (const float*)d_in[14]};
    const float* Wl     = (const float*)d_in[15];
    const float* bl     = (const float*)d_in[16];
    float*       out    = (float*)d_out;

    // workspace layout
    char*  ws  = (char*)d_ws;
    size_t off = 0;
    float* hbuf   = (float*)(ws + off); off += align_up((size_t)N_NODES * FDIM * 4, 256);
    float* aggbuf = (float*)(ws + off); off += align_up((size_t)N_NODES * FDIM * 4, 256);
    float* asrc   = (float*)(ws + off); off += align_up((size_t)N_NODES * 4, 256);
    float* adst   = (float*)(ws + off); off += align_up((size_t)N_NODES * 4, 256);
    float* mbuf   = (float*)(ws + off); off += align_up((size_t)N_NODES * 4, 256);
    float* denom  = (float*)(ws + off); off += align_up((size_t)N_NODES * 4, 256);
    float* escr   = (float*)(ws + off); off += align_up((size_t)E_TOT * 4, 256);
    float* pooled = (float*)(ws + off); off += align_up((size_t)N_GRAPHS * FDIM * 4, 256);
    float* counts = (float*)(ws + off); off += align_up((size_t)N_GRAPHS * 4, 256);

    const int B = 256;
    const int gGemm  = N_NODES / 16;                              // 3125 (exact)
    const int gAlpha = (N_NODES * 32 + B - 1) / B;
    const int gNode  = (N_NODES + B - 1) / B;
    const int gFeat  = (N_NODES * FDIM + B - 1) / B;
    const int gEdge  = (E_TOT + B - 1) / B;
    const int gEAgg  = (int)(((long long)E_TOT * 32 + B - 1) / B);
    const int gPool  = (int)(((long long)N_NODES * 32 + B - 1) / B);
    const int gFin   = (N_GRAPHS * 32 + B - 1) / B;

    const float* cur = x;                     // layer input
    for (int l = 0; l < 3; ++l) {
        // h = maybe_relu(cur) @ W_l   (ReLU of previous layer fused on load)
        gemm_wmma<<<gGemm, B, 0, stream>>>(cur, Wm[l], hbuf, l > 0 ? 1 : 0);
        // per-node attention logits
        alpha_dots<<<gAlpha, B, 0, stream>>>(hbuf, avs[l], avd[l], asrc, adst);
        // softmax scratch init
        fill_f32<<<gNode, B, 0, stream>>>(mbuf,  -INFINITY, N_NODES);
        fill_f32<<<gNode, B, 0, stream>>>(denom, 0.0f,      N_NODES);
        bias_init<<<gFeat, B, 0, stream>>>(aggbuf, bv[l]);
        // segment softmax + weighted scatter
        edge_score<<<gEdge, B, 0, stream>>>(esrc, edst, asrc, adst, escr, mbuf);
        edge_exp  <<<gEdge, B, 0, stream>>>(esrc, edst, escr, mbuf, denom);
        edge_agg  <<<gEAgg, B, 0, stream>>>(esrc, edst, hbuf, escr, denom, aggbuf);
        cur = aggbuf;                         // next layer reads aggregated output
    }

    // global mean pool + final linear
    fill_f32<<<(N_GRAPHS * FDIM + B - 1) / B, B, 0, stream>>>(pooled, 0.0f, N_GRAPHS * FDIM);
    fill_f32<<<(N_GRAPHS + B - 1) / B,        B, 0, stream>>>(counts, 0.0f, N_GRAPHS);
    pool_sum<<<gPool, B, 0, stream>>>(batch, aggbuf, pooled, counts);
    final_linear<<<gFin, B, 0, stream>>>(pooled, counts, Wl, bl, out);
}